// QuantumCircuitSimulator_84396107366855
// MI455X (gfx1250) — compile-verified
//
#include <hip/hip_runtime.h>
#include <stdint.h>

// ---------------------------------------------------------------------------
// 24-qubit state, 8 two-qubit gates with targets (s, s+7), s=0..7.
// All targets < 15 -> each 2^15-amplitude block is closed under the circuit:
// one HBM pass (async B64 stage -> 4 fused LDS passes -> async B64 drain).
// Pass p fuses steps 2p,2p+1 (disjoint qubits -> tensor-product 16x16 complex
// gate), applied as a real 32x32 matrix via chains of V_WMMA_F32_16X16X4_F32.
// LDS uses a GF(2)-linear layout word(a)=XOR Lw[bit]; all addresses are
// maintained with single XORs (Gray-code walks); bank-conflict free by
// construction of Lw; bit0 identity keeps B64 staging contiguous.
// ---------------------------------------------------------------------------

typedef __attribute__((ext_vector_type(2))) float v2f;
typedef __attribute__((ext_vector_type(8))) float v8f;

#define NTHREADS 1024
#define NBLOCKS  512

// Lw[b] = 2^b ^ e[b]; bank parts g[b]=Lw[b]&63 =
// [1,2,4,8,16,32, 32,16,4,2,16,2,4,2,4] -> every per-instruction DS address
// set in every pass spans 5 independent bank bits (32 distinct banks).
static __device__ const uint32_t Lw[15] = {
  1u, 2u, 4u, 8u, 16u, 32u,
  96u,    // 2^6  ^ 32
  144u,   // 2^7  ^ 16
  260u,   // 2^8  ^ 4
  514u,   // 2^9  ^ 2
  1040u,  // 2^10 ^ 16
  2050u,  // 2^11 ^ 2
  4100u,  // 2^12 ^ 4
  8194u,  // 2^13 ^ 2
  16388u  // 2^14 ^ 4
};

__global__ __launch_bounds__(NTHREADS)
void qsim_fused_wmma(const float* __restrict__ state,
                     const float* __restrict__ gates,
                     const long long* __restrict__ targets,
                     float* __restrict__ out)
{
  extern __shared__ __align__(16) float lds[];   // 65536 floats = 256 KB
  const uint32_t tid      = threadIdx.x;
  const uint32_t ldsBase  = (uint32_t)(uintptr_t)lds;
  const uint64_t blockByte = (uint64_t)blockIdx.x * (32768ull * 4ull);

  const uint32_t lane = tid & 31u;
  const uint32_t wav  = tid >> 5;
  const uint32_t n16  = lane & 15u;
  const uint32_t hi   = lane >> 4;

  // ---- stage in: 32x global_load_async_to_lds_b64, Lw-linear LDS layout ----
  uint32_t w0 = (tid * 2u) & 63u;                 // bits 1..5 identity
  if (tid & 32u)  w0 ^= Lw[6];
  if (tid & 64u)  w0 ^= Lw[7];
  if (tid & 128u) w0 ^= Lw[8];
  if (tid & 256u) w0 ^= Lw[9];
  if (tid & 512u) w0 ^= Lw[10];
#pragma unroll
  for (int pl = 0; pl < 2; ++pl) {
    uint64_t sa = (uint64_t)state + ((uint64_t)pl << 26) + blockByte;
#pragma unroll
    for (int s = 0; s < 16; ++s) {
      uint32_t ws = w0 ^ ((s & 1) ? Lw[11] : 0u) ^ ((s & 2) ? Lw[12] : 0u)
                       ^ ((s & 4) ? Lw[13] : 0u) ^ ((s & 8) ? Lw[14] : 0u);
      uint32_t ldsByte = ldsBase + (uint32_t)pl * 131072u + ws * 4u;
      uint32_t voff    = (uint32_t)s * 8192u + tid * 8u;
      asm volatile("global_load_async_to_lds_b64 %0, %1, %2"
                   :: "v"(ldsByte), "v"(voff), "s"(sa) : "memory");
    }
  }
  asm volatile("s_wait_asynccnt 0" ::: "memory");

  // ---- 4 fused passes; k bits (0,1,2,3) <-> qubits (t00,t10,t01,t11) ----
#pragma unroll 1
  for (int p = 0; p < 4; ++p) {
    const int t00 = (int)targets[4 * p + 0];   // step 2p   first target
    const int t01 = (int)targets[4 * p + 1];   // step 2p   second target
    const int t10 = (int)targets[4 * p + 2];   // step 2p+1 first target
    const int t11 = (int)targets[4 * p + 3];   // step 2p+1 second target

    const uint32_t B00w = Lw[t00], B10w = Lw[t10], B01w = Lw[t01], B11w = Lw[t11];

    // sort the 4 targets (network) for free-bit position computation
    int u0 = t00, u1 = t10, u2 = t01, u3 = t11, tw;
    if (u0 > u1) { tw = u0; u0 = u1; u1 = tw; }
    if (u2 > u3) { tw = u2; u2 = u3; u3 = tw; }
    if (u0 > u2) { tw = u0; u0 = u2; u2 = tw; }
    if (u1 > u3) { tw = u1; u1 = u3; u3 = tw; }
    if (u1 > u2) { tw = u1; u1 = u2; u2 = tw; }

    uint32_t C[11];                              // group-bit -> word constant
#pragma unroll
    for (int b = 0; b < 11; ++b) {
      int pos = b;
      pos += (pos >= u0); pos += (pos >= u1); pos += (pos >= u2); pos += (pos >= u3);
      C[b] = Lw[pos];
    }

    // ---- B fragments: fused complex gate F = Gate1 (x) Gate0 (disjoint) ----
    const float* g0 = gates + (2 * p) * 32;
    const float* g1 = gates + (2 * p + 1) * 32;
    const int r0 = 2 * (int)((n16 >> 2) & 1u) + (int)(n16 & 1u);        // Gate0 row
    const int r1 = 2 * (int)((n16 >> 3) & 1u) + (int)((n16 >> 1) & 1u); // Gate1 row
    float g0r[4], g0i[4];
#pragma unroll
    for (int c = 0; c < 4; ++c) { g0r[c] = g0[r0 * 4 + c]; g0i[c] = g0[16 + r0 * 4 + c]; }
    const float g1r0 = g1[r1 * 4 + (int)hi],     g1i0 = g1[16 + r1 * 4 + (int)hi];
    const float g1r2 = g1[r1 * 4 + 2 + (int)hi], g1i2 = g1[16 + r1 * 4 + 2 + (int)hi];

    v2f b0f[8], b1f[8];   // tile0 = out-real, tile1 = out-imag; chains j<4 in-real
#pragma unroll
    for (int jk = 0; jk < 4; ++jk) {
      const int c0 = 2 * (jk & 1);
      const float Gr = (jk & 2) ? g1r2 : g1r0;
      const float Gi = (jk & 2) ? g1i2 : g1i0;
      const float Fr0 = g0r[c0] * Gr - g0i[c0] * Gi;
      const float Fi0 = g0r[c0] * Gi + g0i[c0] * Gr;
      const float Fr1 = g0r[c0 + 1] * Gr - g0i[c0 + 1] * Gi;
      const float Fi1 = g0r[c0 + 1] * Gi + g0i[c0 + 1] * Gr;
      v2f t;
      t.x = Fr0;  t.y = Fr1;  b0f[jk] = t;  b1f[jk + 4] = t;   // +Fr
      t.x = Fi0;  t.y = Fi1;  b1f[jk] = t;                     // +Fi
      t.x = -Fi0; t.y = -Fi1; b0f[jk + 4] = t;                 // -Fi
    }

    // per-lane A-chain offsets (words; +32768 = imag plane)
    uint32_t cj[8];
#pragma unroll
    for (int j = 0; j < 8; ++j) {
      const int jk = j & 3;
      cj[j] = ((jk & 1) ? B01w : 0u) ^ ((jk & 2) ? B11w : 0u)
            ^ (hi ? B10w : 0u) ^ ((j >= 4) ? 32768u : 0u);
    }

    // base word addresses via the linear map (load rows use n16, stores 8*hi)
    const uint32_t Gl = wav * 64u + n16;
    const uint32_t Gs = wav * 64u + 8u * hi;
    uint32_t lb = 0u, sb = 0u;
#pragma unroll
    for (int b = 0; b < 11; ++b) {
      if ((Gl >> b) & 1u) lb ^= C[b];
      if ((Gs >> b) & 1u) sb ^= C[b];
    }
    sb ^= ((n16 & 1u) ? B00w : 0u) ^ ((n16 & 2u) ? B10w : 0u)
        ^ ((n16 & 4u) ? B01w : 0u) ^ ((n16 & 8u) ? B11w : 0u);

    __syncthreads();   // staging / previous pass visible

#pragma unroll
    for (int i = 0; i < 4; ++i) {          // 4 chunks of 16 groups per wave
      v2f af[8];
#pragma unroll
      for (int j = 0; j < 8; ++j) {
        uint32_t t0 = lb ^ cj[j];
        af[j].x = lds[t0];
        af[j].y = lds[t0 ^ B00w];
      }
      v8f d0 = {0.f,0.f,0.f,0.f,0.f,0.f,0.f,0.f};
      v8f d1 = {0.f,0.f,0.f,0.f,0.f,0.f,0.f,0.f};
#pragma unroll
      for (int j = 0; j < 8; ++j) {
        d0 = __builtin_amdgcn_wmma_f32_16x16x4_f32(false, af[j], false, b0f[j],
                                                   (short)0, d0, false, false);
        d1 = __builtin_amdgcn_wmma_f32_16x16x4_f32(false, af[j], false, b1f[j],
                                                   (short)0, d1, false, false);
      }
      {
        uint32_t walk = sb;                // Gray walk over D rows 0,1,3,2,6,7,5,4
        lds[walk] = d0[0];  lds[walk ^ 32768u] = d1[0];  walk ^= C[0];
        lds[walk] = d0[1];  lds[walk ^ 32768u] = d1[1];  walk ^= C[1];
        lds[walk] = d0[3];  lds[walk ^ 32768u] = d1[3];  walk ^= C[0];
        lds[walk] = d0[2];  lds[walk ^ 32768u] = d1[2];  walk ^= C[2];
        lds[walk] = d0[6];  lds[walk ^ 32768u] = d1[6];  walk ^= C[0];
        lds[walk] = d0[7];  lds[walk ^ 32768u] = d1[7];  walk ^= C[1];
        lds[walk] = d0[5];  lds[walk ^ 32768u] = d1[5];  walk ^= C[0];
        lds[walk] = d0[4];  lds[walk ^ 32768u] = d1[4];
      }
      if (i < 3) { uint32_t f = C[(i == 1) ? 5 : 4]; lb ^= f; sb ^= f; }  // chunk Gray
    }
  }

  __syncthreads();

  // ---- drain: 32x global_store_async_from_lds_b64 -> d_out ----
#pragma unroll
  for (int pl = 0; pl < 2; ++pl) {
    uint64_t da = (uint64_t)out + ((uint64_t)pl << 26) + blockByte;
#pragma unroll
    for (int s = 0; s < 16; ++s) {
      uint32_t ws = w0 ^ ((s & 1) ? Lw[11] : 0u) ^ ((s & 2) ? Lw[12] : 0u)
                       ^ ((s & 4) ? Lw[13] : 0u) ^ ((s & 8) ? Lw[14] : 0u);
      uint32_t ldsByte = ldsBase + (uint32_t)pl * 131072u + ws * 4u;
      uint32_t voff    = (uint32_t)s * 8192u + tid * 8u;
      asm volatile("global_store_async_from_lds_b64 %0, %1, %2"
                   :: "v"(voff), "v"(ldsByte), "s"(da) : "memory");
    }
  }
  asm volatile("s_wait_asynccnt 0" ::: "memory");
}

extern "C" void kernel_launch(void* const* d_in, const int* in_sizes, int n_in,
                              void* d_out, int out_size, void* d_ws, size_t ws_size,
                              hipStream_t stream) {
  (void)in_sizes; (void)n_in; (void)out_size; (void)d_ws; (void)ws_size;
  const float*     state   = (const float*)d_in[0];
  const float*     gates   = (const float*)d_in[1];
  const long long* targets = (const long long*)d_in[2];
  float*           out     = (float*)d_out;

  const size_t shmem = 65536u * sizeof(float);   // 262144 B
  (void)hipFuncSetAttribute((const void*)qsim_fused_wmma,
                            hipFuncAttributeMaxDynamicSharedMemorySize,
                            (int)shmem);

  qsim_fused_wmma<<<dim3(NBLOCKS), dim3(NTHREADS), shmem, stream>>>(
      state, gates, targets, out);
}